// CrossAttention_13013750906935
// MI455X (gfx1250) — compile-verified
//
#include <hip/hip_runtime.h>

typedef __attribute__((ext_vector_type(2))) float v2f;
typedef __attribute__((ext_vector_type(8))) float v8f;

#define CC   128   // attention width C == CX
#define SS   64    // text sequence length
#define DSS  512   // text feature dim
#define NPIX 2048  // B*W*H

// ---------------------------------------------------------------------------
// Kernel A: pixel projections via full-fp32 WMMA (V_WMMA_F32_16X16X4_F32).
// Out[p][n] = sum_k X[p][k] * W[k][n] + bias[n] for the 3 weight sets.
// One wave per 16x16 output tile; 32 wmma steps over K=128.
// ---------------------------------------------------------------------------
__global__ __launch_bounds__(32)
void proj_wmma(const float* __restrict__ X,
               const float* __restrict__ Wq, const float* __restrict__ bq,
               const float* __restrict__ Wk, const float* __restrict__ bk,
               const float* __restrict__ Wv, const float* __restrict__ bv,
               float* __restrict__ Qo, float* __restrict__ Ko, float* __restrict__ Vo)
{
    const int lane  = threadIdx.x;       // 0..31
    const int mbase = blockIdx.x * 16;   // pixel-row tile
    const int nbase = blockIdx.y * 16;   // out-channel tile
    const int which = blockIdx.z;

    const float* W    = (which == 0) ? Wq : ((which == 1) ? Wk : Wv);
    const float* bias = (which == 0) ? bq : ((which == 1) ? bk : bv);
    float*       Out  = (which == 0) ? Qo : ((which == 1) ? Ko : Vo);

    const int lm = lane & 15;            // M (for A) / N (for B)
    const int kk = (lane >> 4) << 1;     // lanes 0-15 -> K base 0, lanes 16-31 -> K base 2

    v8f acc = {};
    for (int k = 0; k < CC; k += 4) {
        v2f a, bm;
        const float* ap = X + (size_t)(mbase + lm) * CC + (k + kk);
        a.x = ap[0];                             // A[M=lm][K=k+kk]
        a.y = ap[1];                             // A[M=lm][K=k+kk+1]
        bm.x = W[(size_t)(k + kk) * CC + nbase + lm];       // B[K][N=lm]
        bm.y = W[(size_t)(k + kk + 1) * CC + nbase + lm];
        acc = __builtin_amdgcn_wmma_f32_16x16x4_f32(
                  false, a, false, bm, (short)0, acc, false, false);
    }

    // D layout: VGPR r -> M = mbase + r + 8*(lane>=16), N = nbase + (lane&15)
    const int n    = nbase + lm;
    const int rofs = (lane >> 4) * 8;
    const float bn = bias[n];
#pragma unroll
    for (int r = 0; r < 8; ++r)
        Out[(size_t)(mbase + rofs + r) * CC + n] = acc[r] + bn;
}

// ---------------------------------------------------------------------------
// Kernel O: final projection via WMMA with fused bias + residual:
// out[p][n] = sum_d A[p][d] * Wo[d][n] + bo[n] + x[p][n]
// ---------------------------------------------------------------------------
__global__ __launch_bounds__(32)
void proj_out_wmma(const float* __restrict__ A,
                   const float* __restrict__ Wo, const float* __restrict__ bo,
                   const float* __restrict__ x, float* __restrict__ out)
{
    const int lane  = threadIdx.x;
    const int mbase = blockIdx.x * 16;
    const int nbase = blockIdx.y * 16;

    const int lm = lane & 15;
    const int kk = (lane >> 4) << 1;

    v8f acc = {};
    for (int k = 0; k < CC; k += 4) {
        v2f a, bm;
        const float* ap = A + (size_t)(mbase + lm) * CC + (k + kk);
        a.x = ap[0];
        a.y = ap[1];
        bm.x = Wo[(size_t)(k + kk) * CC + nbase + lm];
        bm.y = Wo[(size_t)(k + kk + 1) * CC + nbase + lm];
        acc = __builtin_amdgcn_wmma_f32_16x16x4_f32(
                  false, a, false, bm, (short)0, acc, false, false);
    }

    const int n    = nbase + lm;
    const int rofs = (lane >> 4) * 8;
    const float bn = bo[n];
#pragma unroll
    for (int r = 0; r < 8; ++r) {
        const size_t idx = (size_t)(mbase + rofs + r) * CC + n;
        out[idx] = acc[r] + bn + x[idx];
    }
}

// ---------------------------------------------------------------------------
// Kernel S1: text projections Ks = s@Wks+bks, Vs = s@Wvs+bvs  (tiny GEMMs)
// grid (32, batch, which), block 256: one output element per thread.
// ---------------------------------------------------------------------------
__global__ __launch_bounds__(256)
void sproj(const float* __restrict__ s,
           const float* __restrict__ Wks, const float* __restrict__ bks,
           const float* __restrict__ Wvs, const float* __restrict__ bvs,
           float* __restrict__ Ks, float* __restrict__ Vs)
{
    const int idx  = blockIdx.x * 256 + threadIdx.x;   // 0..8191
    const int b    = blockIdx.y;
    const int m    = blockIdx.z;
    const int srow = idx >> 7;
    const int d    = idx & 127;

    const float* W    = m ? Wvs : Wks;
    const float* bias = m ? bvs : bks;
    float*       Out  = m ? Vs : Ks;

    const float* srp = s + ((size_t)b * SS + srow) * DSS;
    float a = bias[d];
    for (int i = 0; i < DSS; ++i)
        a += srp[i] * W[(size_t)i * CC + d];
    Out[((size_t)b * SS + srow) * CC + d] = a;
}

// ---------------------------------------------------------------------------
// Kernel S2: per-batch Gram Gs = Vs^T Ks  [128x128], plus column sums of Ks/Vs.
// grid (128 rows, batch), block 128 (one d per thread).
// ---------------------------------------------------------------------------
__global__ __launch_bounds__(128)
void gram_s(const float* __restrict__ Ks, const float* __restrict__ Vs,
            float* __restrict__ Gs, float* __restrict__ sumK, float* __restrict__ sumV)
{
    const int c = blockIdx.x;
    const int b = blockIdx.y;
    const int d = threadIdx.x;
    const float* Kb = Ks + (size_t)b * SS * CC;
    const float* Vb = Vs + (size_t)b * SS * CC;

    float a = 0.f;
    for (int sIt = 0; sIt < SS; ++sIt)
        a += Vb[sIt * CC + c] * Kb[sIt * CC + d];
    Gs[((size_t)b * CC + c) * CC + d] = a;

    if (c == 0) {
        float sk = 0.f;
        for (int sIt = 0; sIt < SS; ++sIt) sk += Kb[sIt * CC + d];
        sumK[b * CC + d] = sk;
    }
    if (c == 1) {
        float sv = 0.f;
        for (int sIt = 0; sIt < SS; ++sIt) sv += Vb[sIt * CC + d];
        sumV[b * CC + d] = sv;
    }
}

// ---------------------------------------------------------------------------
// Kernel B: per-pixel attention core. One 256-thread block (8 wave32)/pixel.
// Thread t = (c = t>>1, half = t&1) owns 64 d-values of attention row c.
// G[c,d] = Gs[c,d] + 64*vx[c]*kx[d] + vx[c]*sumK[d] + sumV[c]*kx[d]  (exact)
// Writes the per-pixel vector  a[d] = sum_c (q_c/Z_c) e_{c,d}  to Aout;
// the Wo projection + bias + residual is done by proj_out_wmma.
// ---------------------------------------------------------------------------
__global__ __launch_bounds__(256)
void attn_pixel(const float* __restrict__ Q, const float* __restrict__ KX,
                const float* __restrict__ VX,
                const float* __restrict__ Gs, const float* __restrict__ sumK,
                const float* __restrict__ sumV,
                float* __restrict__ Aout)
{
    __shared__ float s_kx[CC], s_vx[CC], s_q[CC], s_sk[CC], s_sv[CC], s_out[CC];

    const int p = blockIdx.x;
    const int b = p >> 10;            // p / (W*H) = p / 1024
    const int t = threadIdx.x;

    if (t < CC) {
        s_kx[t] = KX[(size_t)p * CC + t];
        s_vx[t] = VX[(size_t)p * CC + t];
        s_q[t]  = Q [(size_t)p * CC + t];
        s_sk[t] = sumK[b * CC + t];
        s_sv[t] = sumV[b * CC + t];
        s_out[t] = 0.f;
    }
    __syncthreads();

    const int   c     = t >> 1;
    const int   half  = t & 1;
    const int   dbase = half << 6;
    const float vc    = s_vx[c];
    const float svc   = s_sv[c];
    const float scale = 0.08838834764831843f;   // 1/sqrt(128)
    const float* gsr  = Gs + ((size_t)b * CC + c) * CC + dbase;

    float gv[64];
    float mx = -3.402823466e38f;
#pragma unroll
    for (int j = 0; j < 64; ++j) {
        const int d = dbase + j;
        const float kd = s_kx[d];
        const float g = (gsr[j] + vc * (64.0f * kd + s_sk[d]) + svc * kd) * scale;
        gv[j] = g;
        mx = fmaxf(mx, g);
    }
    mx = fmaxf(mx, __shfl_xor(mx, 1, 32));      // combine the two halves of row c

    float z = 0.f;
#pragma unroll
    for (int j = 0; j < 64; ++j) {
        const float e = __expf(gv[j] - mx);
        gv[j] = e;
        z += e;
    }
    z += __shfl_xor(z, 1, 32);

    const float w = s_q[c] / z;                 // q[c] / Z_c
    const bool leader = ((t & 31) < 2);         // lanes 0 (half 0) and 1 (half 1)

    // a[d] += sum_c w_c * e_{c,d}: butterfly over the 16 same-half lanes,
    // then 2 atomics per wave (8-way cross-wave accumulation in LDS).
#pragma unroll
    for (int j = 0; j < 64; ++j) {
        float v = w * gv[j];
        v += __shfl_xor(v, 2, 32);
        v += __shfl_xor(v, 4, 32);
        v += __shfl_xor(v, 8, 32);
        v += __shfl_xor(v, 16, 32);
        if (leader) atomicAdd(&s_out[dbase + j], v);
    }
    __syncthreads();

    if (t < CC)
        Aout[(size_t)p * CC + t] = s_out[t];
}

// ---------------------------------------------------------------------------
extern "C" void kernel_launch(void* const* d_in, const int* in_sizes, int n_in,
                              void* d_out, int out_size, void* d_ws, size_t ws_size,
                              hipStream_t stream)
{
    (void)in_sizes; (void)n_in; (void)out_size; (void)ws_size;

    const float* x   = (const float*)d_in[0];
    const float* s   = (const float*)d_in[1];
    const float* Wq  = (const float*)d_in[2];
    const float* bq  = (const float*)d_in[3];
    const float* Wkx = (const float*)d_in[4];
    const float* bkx = (const float*)d_in[5];
    const float* Wvx = (const float*)d_in[6];
    const float* bvx = (const float*)d_in[7];
    const float* Wks = (const float*)d_in[8];
    const float* bks = (const float*)d_in[9];
    const float* Wvs = (const float*)d_in[10];
    const float* bvs = (const float*)d_in[11];
    const float* Wo  = (const float*)d_in[12];
    const float* bo  = (const float*)d_in[13];
    float* out = (float*)d_out;

    float* ws  = (float*)d_ws;
    float* Qw  = ws;                          // 2048*128
    float* KXw = Qw  + (size_t)NPIX * CC;     // 2048*128
    float* VXw = KXw + (size_t)NPIX * CC;     // 2048*128
    float* Aw  = VXw + (size_t)NPIX * CC;     // 2048*128 (post-softmax q@attn)
    float* Ksw = Aw  + (size_t)NPIX * CC;     // 2*64*128
    float* Vsw = Ksw + (size_t)2 * SS * CC;   // 2*64*128
    float* Gsw = Vsw + (size_t)2 * SS * CC;   // 2*128*128
    float* sKw = Gsw + (size_t)2 * CC * CC;   // 2*128
    float* sVw = sKw + (size_t)2 * CC;        // 2*128

    proj_wmma<<<dim3(NPIX / 16, CC / 16, 3), 32, 0, stream>>>(
        x, Wq, bq, Wkx, bkx, Wvx, bvx, Qw, KXw, VXw);
    sproj<<<dim3(32, 2, 2), 256, 0, stream>>>(
        s, Wks, bks, Wvs, bvs, Ksw, Vsw);
    gram_s<<<dim3(CC, 2), 128, 0, stream>>>(
        Ksw, Vsw, Gsw, sKw, sVw);
    attn_pixel<<<NPIX, 256, 0, stream>>>(
        Qw, KXw, VXw, Gsw, sKw, sVw, Aw);
    proj_out_wmma<<<dim3(NPIX / 16, CC / 16), 32, 0, stream>>>(
        Aw, Wo, bo, x, out);
}